// mLowRNorm_20366734917810
// MI455X (gfx1250) — compile-verified
//
#include <hip/hip_runtime.h>
#include <math.h>

typedef __attribute__((ext_vector_type(2))) float v2f;
typedef __attribute__((ext_vector_type(8))) float v8f;

#define T_FULL 262144
#define D_FULL 256
#define H_CH   32
#define L_F    41
#define TZ     (T_FULL - L_F + 1)   // 262104
#define TD     (TZ - L_F + 1)       // 262064
#define FILT_N (H_CH * L_F)         // 1312 floats per bank

static __device__ __forceinline__ v8f wmma_f32_k4(v2f a, v2f b, v8f c) {
  // V_WMMA_F32_16X16X4_F32 : D = A(16x4,f32) * B(4x16,f32) + C(16x16,f32)
  return __builtin_amdgcn_wmma_f32_16x16x4_f32(
      /*neg_a=*/false, a, /*neg_b=*/false, b,
      /*c_mod=*/(short)0, c, /*reuse_a=*/false, /*reuse_b=*/false);
}

// ---------------------------------------------------------------------------
// Kernel 1: Gaussian filter banks (enc: -mu, dec: +mu) + sphere-normalized W2
// ---------------------------------------------------------------------------
__global__ void setup_kernel(const float* __restrict__ mus,
                             const float* __restrict__ sigmas,
                             const float* __restrict__ scaling,
                             const float* __restrict__ V2,
                             float* __restrict__ filt,   // [4][H][L]
                             float* __restrict__ w2n) {  // [256][32]
  const int tid = threadIdx.x;
  for (int idx = tid; idx < 4 * FILT_N; idx += blockDim.x) {
    const int f   = idx / FILT_N;          // 0:enc r0, 1:enc r1, 2:dec r0, 3:dec r1
    const int rem = idx - f * FILT_N;
    const int h   = rem / L_F;
    const int l   = rem - h * L_F;
    const int r   = f & 1;
    float mu = mus[r * H_CH + h];
    if (f < 2) mu = -mu;                   // encode banks use flipped mu
    float sig = sigmas[r * H_CH + h];
    sig = sig < 1.0f ? 1.0f : sig;         // clamp(min=1)
    const float amp = scaling[r * H_CH + h];
    const float x = (float)(l - 21);       // XVALS = -21..19
    const float u = (x - mu) / sig;
    filt[idx] = amp * expf(-0.5f * u * u);
  }
  if (tid < H_CH) {
    float s = 0.0f;
    for (int d = 0; d < D_FULL; ++d) {
      const float v = V2[d * H_CH + tid];
      s += v * v;
    }
    const float inv = rsqrtf(s);
    for (int d = 0; d < D_FULL; ++d)
      w2n[d * H_CH + tid] = V2[d * H_CH + tid] * inv;
  }
}

// ---------------------------------------------------------------------------
// Kernel 2: z{0,1}[t,h] = x{0,1}[t,:] . W1[h, half*128 : half*128+128]
// fp32 WMMA GEMM: per wave 16 M-rows x 32 N, K=128 -> 64 v_wmma per wave.
// ---------------------------------------------------------------------------
__global__ void __launch_bounds__(256)
proj_kernel(const float* __restrict__ x0, const float* __restrict__ x1,
            const float* __restrict__ W1,
            float* __restrict__ z0ws, float* __restrict__ z1ws) {
  const int lane = threadIdx.x & 31;
  const int wave = threadIdx.x >> 5;
  const int lo   = lane & 15;
  const int hi   = lane >> 4;
  const int m0   = blockIdx.x * 128 + wave * 16;
  const int reg  = blockIdx.y;

  const float* __restrict__ X  = reg ? x1 : x0;
  float* __restrict__       Z  = reg ? z1ws : z0ws;
  const int wk0 = reg * 128;   // K-offset inside W1 rows (row stride 256)

  v8f c0 = {0.f,0.f,0.f,0.f,0.f,0.f,0.f,0.f};
  v8f c1 = {0.f,0.f,0.f,0.f,0.f,0.f,0.f,0.f};

  // A: lane<16 holds K={k,k+1} of row m0+lo; lane>=16 holds K={k+2,k+3}
  const float* arow  = X  + (size_t)(m0 + lo) * 128 + 2 * hi;
  // B[kk][n] = W1[n][k+kk]; same half-lane K split as A
  const float* b0row = W1 + (size_t)(lo)      * 256 + wk0 + 2 * hi;
  const float* b1row = W1 + (size_t)(16 + lo) * 256 + wk0 + 2 * hi;

  for (int k = 0; k < 128; k += 4) {
    const v2f a  = *(const v2f*)(arow  + k);
    const v2f b0 = *(const v2f*)(b0row + k);
    const v2f b1 = *(const v2f*)(b1row + k);
    c0 = wmma_f32_k4(a, b0, c0);
    c1 = wmma_f32_k4(a, b1, c1);
  }

  // C/D: VGPR v -> M = v + 8*hi, N = lo
#pragma unroll
  for (int v = 0; v < 8; ++v) {
    const int row = m0 + v + 8 * hi;
    Z[(size_t)row * 32 + lo]      = c0[v];
    Z[(size_t)row * 32 + 16 + lo] = c1[v];
  }
}

// ---------------------------------------------------------------------------
// Kernel 3: encode depthwise conv:  z[t,h] = b1[h] + sum_l f0[h,l]*z0[t+l,h]
//                                                  + f1[h,l]*z1[t+l,h]
// 128 t-rows per block, 168-row input tile staged in LDS (lane==channel ->
// conflict-free banks).
// ---------------------------------------------------------------------------
__global__ void __launch_bounds__(256)
enc_conv_kernel(const float* __restrict__ z0ws, const float* __restrict__ z1ws,
                const float* __restrict__ filt, const float* __restrict__ b1,
                float* __restrict__ zws) {
  __shared__ float s0[168 * 32];
  __shared__ float s1[168 * 32];
  __shared__ float f0[FILT_N];
  __shared__ float f1[FILT_N];

  const int tid = threadIdx.x;
  const int t0  = blockIdx.x * 128;

  for (int i = tid; i < 168 * 32; i += 256) {
    const int row = t0 + (i >> 5);
    float a = 0.0f, b = 0.0f;
    if (row < T_FULL) {
      a = z0ws[(size_t)row * 32 + (i & 31)];
      b = z1ws[(size_t)row * 32 + (i & 31)];
    }
    s0[i] = a;
    s1[i] = b;
  }
  for (int i = tid; i < FILT_N; i += 256) {
    f0[i] = filt[i];
    f1[i] = filt[FILT_N + i];
  }
  __syncthreads();

  const int h  = tid & 31;
  const int tq = tid >> 5;          // wave id 0..7
  const float bb = b1[h];
  for (int j = 0; j < 16; ++j) {
    const int tl = tq * 16 + j;
    float acc = bb;
#pragma unroll
    for (int l = 0; l < L_F; ++l) {
      acc = fmaf(f0[h * L_F + l], s0[(tl + l) * 32 + h], acc);
      acc = fmaf(f1[h * L_F + l], s1[(tl + l) * 32 + h], acc);
    }
    const int t = t0 + tl;
    if (t < TZ) zws[(size_t)t * 32 + h] = acc;
  }
}

// ---------------------------------------------------------------------------
// Kernel 4: decode depthwise convs -> zd0, zd1 (first two outputs)
// ---------------------------------------------------------------------------
__global__ void __launch_bounds__(256)
dec_conv_kernel(const float* __restrict__ zws, const float* __restrict__ filt,
                float* __restrict__ out) {
  __shared__ float sz[168 * 32];
  __shared__ float f2[FILT_N];
  __shared__ float f3[FILT_N];

  const int tid = threadIdx.x;
  const int t0  = blockIdx.x * 128;

  for (int i = tid; i < 168 * 32; i += 256) {
    const int row = t0 + (i >> 5);
    sz[i] = (row < TZ) ? zws[(size_t)row * 32 + (i & 31)] : 0.0f;
  }
  for (int i = tid; i < FILT_N; i += 256) {
    f2[i] = filt[2 * FILT_N + i];
    f3[i] = filt[3 * FILT_N + i];
  }
  __syncthreads();

  const int h  = tid & 31;
  const int tq = tid >> 5;
  float* __restrict__ zd0 = out;
  float* __restrict__ zd1 = out + (size_t)TD * 32;

  for (int j = 0; j < 16; ++j) {
    const int tl = tq * 16 + j;
    float a0 = 0.0f, a1 = 0.0f;
#pragma unroll
    for (int l = 0; l < L_F; ++l) {
      const float zv = sz[(tl + l) * 32 + h];
      a0 = fmaf(f2[h * L_F + l], zv, a0);
      a1 = fmaf(f3[h * L_F + l], zv, a1);
    }
    const int t = t0 + tl;
    if (t < TD) {
      zd0[(size_t)t * 32 + h] = a0;
      zd1[(size_t)t * 32 + h] = a1;
    }
  }
}

// ---------------------------------------------------------------------------
// Kernel 5: xh = zd @ W2half.T + b2half.  M=TD, K=32, N=128.
// Per wave: 16 M-rows x full 128 N (8 accumulators), 64 v_wmma per wave.
// ---------------------------------------------------------------------------
__global__ void __launch_bounds__(256)
dec_gemm_kernel(const float* __restrict__ outbase, const float* __restrict__ w2n,
                const float* __restrict__ b2, float* __restrict__ out) {
  const int lane = threadIdx.x & 31;
  const int wave = threadIdx.x >> 5;
  const int lo   = lane & 15;
  const int hi   = lane >> 4;
  const int half = blockIdx.y;  // 0: zd0->xh0, 1: zd1->xh1

  const float* __restrict__ ZD   = outbase + (size_t)half * TD * 32;
  const float* __restrict__ Bp   = w2n + (size_t)half * 128 * 32;   // W2n[n][h]
  const float* __restrict__ bias = b2 + half * 128;
  float* __restrict__ XH = out + (size_t)2 * TD * 32 + (size_t)half * TD * 128;

  const int m0 = blockIdx.x * 128 + wave * 16;
  int mr = m0 + lo;
  mr = mr < (TD - 1) ? mr : (TD - 1);  // clamp tail loads (lane select, no branch)
  const float* arow = ZD + (size_t)mr * 32 + 2 * hi;

  v8f c[8];
#pragma unroll
  for (int nt = 0; nt < 8; ++nt) c[nt] = (v8f){0.f,0.f,0.f,0.f,0.f,0.f,0.f,0.f};

  for (int k = 0; k < 32; k += 4) {
    const v2f a = *(const v2f*)(arow + k);
#pragma unroll
    for (int nt = 0; nt < 8; ++nt) {
      const v2f b = *(const v2f*)(Bp + (size_t)(nt * 16 + lo) * 32 + k + 2 * hi);
      c[nt] = wmma_f32_k4(a, b, c[nt]);
    }
  }

#pragma unroll
  for (int nt = 0; nt < 8; ++nt) {
    const float bv = bias[nt * 16 + lo];
#pragma unroll
    for (int v = 0; v < 8; ++v) {
      const int row = m0 + v + 8 * hi;
      if (row < TD)
        __builtin_nontemporal_store(c[nt][v] + bv,
                                    &XH[(size_t)row * 128 + nt * 16 + lo]);
    }
  }
}

// ---------------------------------------------------------------------------
extern "C" void kernel_launch(void* const* d_in, const int* in_sizes, int n_in,
                              void* d_out, int out_size, void* d_ws, size_t ws_size,
                              hipStream_t stream) {
  const float* x0      = (const float*)d_in[0];
  const float* x1      = (const float*)d_in[1];
  const float* W1      = (const float*)d_in[2];
  const float* b1      = (const float*)d_in[3];
  const float* V2      = (const float*)d_in[4];
  const float* b2      = (const float*)d_in[5];
  const float* mus     = (const float*)d_in[6];
  const float* sigmas  = (const float*)d_in[7];
  const float* scaling = (const float*)d_in[8];
  float* out = (float*)d_out;
  float* ws  = (float*)d_ws;

  const size_t SZ = (size_t)T_FULL * 32;   // 8,388,608 floats
  float* z0ws = ws;                        // [T, 32]
  float* z1ws = ws + SZ;                   // [T, 32]
  float* zws  = ws + 2 * SZ;               // [TZ, 32]
  float* filt = ws + 3 * SZ;               // [4][32][41]
  float* w2n  = filt + 4 * FILT_N;         // [256][32]

  setup_kernel<<<1, 256, 0, stream>>>(mus, sigmas, scaling, V2, filt, w2n);

  proj_kernel<<<dim3(T_FULL / 128, 2), 256, 0, stream>>>(x0, x1, W1, z0ws, z1ws);

  enc_conv_kernel<<<(TZ + 127) / 128, 256, 0, stream>>>(z0ws, z1ws, filt, b1, zws);

  dec_conv_kernel<<<(TD + 127) / 128, 256, 0, stream>>>(zws, filt, out);

  dec_gemm_kernel<<<dim3((TD + 127) / 128, 2), 256, 0, stream>>>(out, w2n, b2, out);
}